// ClosestETFGeometryLayer_30597347017000
// MI455X (gfx1250) — compile-verified
//
#include <hip/hip_runtime.h>

typedef float v2f __attribute__((ext_vector_type(2)));
typedef float v8f __attribute__((ext_vector_type(8)));

#define D_   256
#define KA_  50
#define KP_  64
#define NT_  512

__device__ __forceinline__ v8f wmma_f32(v2f a, v2f b, v8f c) {
  // V_WMMA_F32_16X16X4_F32: (neg_a, A, neg_b, B, c_mod, C, reuse_a, reuse_b)
  return __builtin_amdgcn_wmma_f32_16x16x4_f32(false, a, false, b, (short)0, c,
                                               false, false);
}

__global__ __launch_bounds__(NT_) void etf_rgd_kernel(
    const float* __restrict__ Y, const float* __restrict__ Min,
    float* __restrict__ ym, float* __restrict__ prox, float* __restrict__ out) {
  __shared__ float Xs[D_ * KP_];   // 64 KB  current X (padded cols 50..63 = 0)
  __shared__ float Gs[D_ * KP_];   // 64 KB  gradient / temp Y
  __shared__ float Ms[KP_ * KP_];  // 16 KB  centering matrix, padded
  __shared__ float Ss[KP_ * KP_];  // 16 KB  X^T G / sym / MGS scratch

  const int t = threadIdx.x;
  const int lane = t & 31;
  const int wv = t >> 5;
  const int m16 = lane & 15;
  const int koff = (lane >> 4) << 1;  // K offset 0 or 2 (A/B fragment halves)
  const int row8 = (lane >> 4) << 3;  // C/D row offset 0 or 8

  const float c1 = 2.0f / (KA_ - 1);
  const float c2 = 2.0f / 7.0f;  // 2/sqrt(K-1), K=50
  const float delta = 1e-3f;
  const float lr = 0.5f / (c1 + delta);

  // ---- load M (padded with zeros) and Y (into Gs, padded) ----
  for (int e = t; e < KP_ * KP_; e += NT_) {
    int r = e >> 6, c = e & 63;
    Ms[e] = (r < KA_ && c < KA_) ? Min[r * KA_ + c] : 0.0f;
  }
  for (int e = t; e < D_ * KP_; e += NT_) {
    int r = e >> 6, c = e & 63;
    Gs[e] = (c < KA_) ? Y[r * KA_ + c] : 0.0f;
  }
  __syncthreads();

  // ---- YM = Y @ M  (once), store to global scratch ----
  {
    const int r0 = wv << 4;
    for (int ct = 0; ct < 4; ++ct) {
      v8f acc = {};
#pragma unroll
      for (int kt = 0; kt < 16; ++kt) {
        int k = kt << 2;
        v2f a, b;
        a.x = Gs[(r0 + m16) * KP_ + k + koff];
        a.y = Gs[(r0 + m16) * KP_ + k + koff + 1];
        b.x = Ms[(k + koff) * KP_ + (ct << 4) + m16];
        b.y = Ms[(k + koff + 1) * KP_ + (ct << 4) + m16];
        acc = wmma_f32(a, b, acc);
      }
#pragma unroll
      for (int v = 0; v < 8; ++v)
        ym[(r0 + row8 + v) * KP_ + (ct << 4) + m16] = acc[v];
    }
  }
  __threadfence();
  __syncthreads();

  // ---- X0 = eye(256, 64) truncated to 50 cols ----
  for (int e = t; e < D_ * KP_; e += NT_) {
    int r = e >> 6, c = e & 63;
    Xs[e] = (r == c && c < KA_) ? 1.0f : 0.0f;
  }
  __syncthreads();

#pragma clang loop unroll(disable)
  for (int phase = 0; phase < 2; ++phase) {
    if (phase == 1) {  // Prox := P_feas (result of phase 0)
      for (int e = t; e < D_ * KP_; e += NT_) prox[e] = Xs[e];
      __threadfence();
      __syncthreads();
    }
#pragma clang loop unroll(disable)
    for (int it = 0; it < 500; ++it) {
      // ---- G = c1*(X@M) - c2*YM  (+ delta*(X - Prox) in phase 1) ----
      {
        const int r0 = wv << 4;
#pragma clang loop unroll(disable)
        for (int ct = 0; ct < 4; ++ct) {
          v8f acc = {};
#pragma unroll
          for (int kt = 0; kt < 16; ++kt) {
            int k = kt << 2;
            v2f a, b;
            a.x = Xs[(r0 + m16) * KP_ + k + koff];
            a.y = Xs[(r0 + m16) * KP_ + k + koff + 1];
            b.x = Ms[(k + koff) * KP_ + (ct << 4) + m16];
            b.y = Ms[(k + koff + 1) * KP_ + (ct << 4) + m16];
            acc = wmma_f32(a, b, acc);
          }
#pragma unroll
          for (int v = 0; v < 8; ++v) {
            int idx = (r0 + row8 + v) * KP_ + (ct << 4) + m16;
            float g = c1 * acc[v] - c2 * ym[idx];
            if (phase == 1) g += delta * (Xs[idx] - prox[idx]);
            Gs[idx] = g;
          }
        }
      }
      __syncthreads();
      // ---- S = X^T @ G  (16 tiles, one per wave, inner dim 256) ----
      {
        const int r0 = (wv >> 2) << 4;
        const int c0 = (wv & 3) << 4;
        v8f acc = {};
#pragma unroll
        for (int kt = 0; kt < 64; ++kt) {
          int k = kt << 2;
          v2f a, b;
          a.x = Xs[(k + koff) * KP_ + r0 + m16];      // (X^T)[m][k] = X[k][m]
          a.y = Xs[(k + koff + 1) * KP_ + r0 + m16];
          b.x = Gs[(k + koff) * KP_ + c0 + m16];
          b.y = Gs[(k + koff + 1) * KP_ + c0 + m16];
          acc = wmma_f32(a, b, acc);
        }
#pragma unroll
        for (int v = 0; v < 8; ++v)
          Ss[(r0 + row8 + v) * KP_ + c0 + m16] = acc[v];
      }
      __syncthreads();
      // ---- S = 0.5*(S + S^T) ----
      {
        float sv[8];
#pragma unroll
        for (int p = 0; p < 8; ++p) {
          int e = t + p * NT_;
          int i = e >> 6, j2 = e & 63;
          sv[p] = 0.5f * (Ss[i * KP_ + j2] + Ss[j2 * KP_ + i]);
        }
        __syncthreads();
#pragma unroll
        for (int p = 0; p < 8; ++p) Ss[t + p * NT_] = sv[p];
      }
      __syncthreads();
      // ---- rg = G - X @ Ssym (in place in Gs) ----
      {
        const int r0 = wv << 4;
#pragma clang loop unroll(disable)
        for (int ct = 0; ct < 4; ++ct) {
          v8f acc = {};
#pragma unroll
          for (int kt = 0; kt < 16; ++kt) {
            int k = kt << 2;
            v2f a, b;
            a.x = Xs[(r0 + m16) * KP_ + k + koff];
            a.y = Xs[(r0 + m16) * KP_ + k + koff + 1];
            b.x = Ss[(k + koff) * KP_ + (ct << 4) + m16];
            b.y = Ss[(k + koff + 1) * KP_ + (ct << 4) + m16];
            acc = wmma_f32(a, b, acc);
          }
#pragma unroll
          for (int v = 0; v < 8; ++v) {
            int idx = (r0 + row8 + v) * KP_ + (ct << 4) + m16;
            Gs[idx] = Gs[idx] - acc[v];
          }
        }
      }
      __syncthreads();
      // ---- A = X - lr*rg ----
      for (int e = t; e < D_ * KP_; e += NT_) Xs[e] -= lr * Gs[e];
      __syncthreads();
      // ---- QR retraction via modified Gram-Schmidt (R diag > 0 == _qf) ----
      float* scratch = Ss;        // Ss contents dead now; reuse 4096 floats
      float* dots = Ss + 2048;
#pragma clang loop unroll(disable)
      for (int j = 0; j < KA_; ++j) {
        float p = 0.0f;
        if (t < D_) {
          float v = Xs[t * KP_ + j];
          p = v * v;
        }
        scratch[t] = p;
        __syncthreads();
#pragma clang loop unroll(disable)
        for (int s = NT_ >> 1; s > 0; s >>= 1) {
          if (t < s) scratch[t] += scratch[t + s];
          __syncthreads();
        }
        float inv = __frsqrt_rn(scratch[0]);
        if (t < D_) Xs[t * KP_ + j] *= inv;
        __syncthreads();
        // dots of unit column j with all later columns (one column per wave)
#pragma clang loop unroll(disable)
        for (int rnd = 0; rnd < 4; ++rnd) {
          int c = j + 1 + wv + (rnd << 4);
          if (c < KA_) {
            float pp = 0.0f;
#pragma unroll
            for (int r = lane; r < D_; r += 32)
              pp += Xs[r * KP_ + c] * Xs[r * KP_ + j];
            scratch[c * 32 + lane] = pp;
          }
        }
        __syncthreads();
        if (t < KP_) {
          int c = t;
          if (c > j && c < KA_) {
            float s = 0.0f;
#pragma unroll
            for (int l = 0; l < 32; ++l) s += scratch[c * 32 + l];
            dots[c] = s;
          }
        }
        __syncthreads();
        int nc = KA_ - 1 - j;
        for (int e = t; e < D_ * nc; e += NT_) {
          int r = e / nc;
          int c = j + 1 + (e - r * nc);
          Xs[r * KP_ + c] -= dots[c] * Xs[r * KP_ + j];
        }
        __syncthreads();
      }
    }
  }
  // ---- write P (256 x 50, row-major) ----
  for (int e = t; e < D_ * KA_; e += NT_) {
    int r = e / KA_, c = e - r * KA_;
    out[e] = Xs[r * KP_ + c];
  }
}

extern "C" void kernel_launch(void* const* d_in, const int* in_sizes, int n_in,
                              void* d_out, int out_size, void* d_ws,
                              size_t ws_size, hipStream_t stream) {
  (void)in_sizes; (void)n_in; (void)out_size; (void)ws_size;
  const float* Y = (const float*)d_in[0];    // feature_means (256 x 50)
  const float* M = (const float*)d_in[1];    // centering matrix (50 x 50)
  // d_in[2] (LCI) only feeds the backward pass; unused in forward.
  float* ym = (float*)d_ws;                  // 256*64 floats (64 KB)
  float* prox = ym + D_ * KP_;               // 256*64 floats (64 KB)
  float* out = (float*)d_out;                // P (256 x 50)
  etf_rgd_kernel<<<dim3(1), dim3(NT_), 0, stream>>>(Y, M, ym, prox, out);
}